// RSSM_55722905699145
// MI455X (gfx1250) — compile-verified
//
#include <hip/hip_runtime.h>
#include <math.h>

typedef __attribute__((ext_vector_type(2))) float v2f;
typedef __attribute__((ext_vector_type(8))) float v8f;

static __device__ __forceinline__ v8f wmma4(v2f a, v2f b, v8f c) {
  // D = A(16x4 f32) x B(4x16 f32) + C(16x16 f32)
  return __builtin_amdgcn_wmma_f32_16x16x4_f32(
      /*neg_a=*/false, a, /*neg_b=*/false, b,
      /*c_mod=*/(short)0, c, /*reuse_a=*/false, /*reuse_b=*/false);
}

// xp/wp already include: row*ld + 2*(lane>>4). Uniform K=1024 partial GEMM.
template <int K>
static __device__ __forceinline__ v8f gemm_k(const float* __restrict__ xp,
                                             const float* __restrict__ wp,
                                             v8f acc) {
#pragma unroll 8
  for (int k = 0; k < K; k += 4) {
    v2f a = *(const v2f*)(xp + k);
    v2f b = *(const v2f*)(wp + k);
    acc = wmma4(a, b, acc);
  }
  return acc;
}

// Workspace layout (floats)
#define WS_GATE   0          // 12 x [16,2048] gate partials
#define WS_HID    393216     // 5 x [16,2048] hidden partials
#define WS_H1PRI  557056     // [16,2048]
#define WS_H1POST 589824     // [16,2048]
#define WS_LOGIT  622592     // 4 x [16,1024] logit partials (pri p0,p1, post p0,p1)

// ---------------- GRU gate partial GEMMs ----------------
// 1536 waves: comp = g*4 + p, g in {r,z,n}, p = K-quarter:
//   p0: prev_stoch x W_ih[:, 0:1024]      p1: action x W_ih[:, 1024:2048]
//   p2: prev_deter[:, 0:1024] x W_hh      p3: prev_deter[:, 1024:2048] x W_hh
__global__ void __launch_bounds__(256) k_gates_gemm(
    const float* __restrict__ prev_stoch, const float* __restrict__ prev_deter,
    const float* __restrict__ action,
    const float* __restrict__ W_ih, const float* __restrict__ W_hh,
    float* __restrict__ ws) {
  const int gid  = blockIdx.x * 8 + (threadIdx.x >> 5);  // 0..1535
  const int comp = gid >> 7;                             // 0..11
  const int tile = gid & 127;
  const int g = comp >> 2, p = comp & 3;
  const int hb = tile * 16;
  const int lane = threadIdx.x & 31;
  const int r16  = lane & 15;
  const int koff = (lane >> 4) * 2;
  const size_t row = (size_t)(g * 2048 + hb + r16);

  const float* xp; const float* wp;
  if (p == 0) {
    xp = prev_stoch + r16 * 1024 + koff;
    wp = W_ih + row * 2048 + koff;
  } else if (p == 1) {
    xp = action + r16 * 1024 + koff;
    wp = W_ih + row * 2048 + 1024 + koff;
  } else if (p == 2) {
    xp = prev_deter + r16 * 2048 + koff;
    wp = W_hh + row * 2048 + koff;
  } else {
    xp = prev_deter + r16 * 2048 + 1024 + koff;
    wp = W_hh + row * 2048 + 1024 + koff;
  }

  v8f acc = {};
  acc = gemm_k<1024>(xp, wp, acc);

  float* dst = ws + WS_GATE + (size_t)comp * 32768;
#pragma unroll
  for (int j = 0; j < 8; ++j) {      // C layout: VGPR j -> M = j + 8*(lane>>4)
    int m = j + 8 * (lane >> 4);
    dst[m * 2048 + hb + r16] = acc[j];
  }
}

// ---------------- GRU combine: fold partials + gates ----------------
__global__ void __launch_bounds__(256) k_gru_combine(
    const float* __restrict__ ws, const float* __restrict__ prev_deter,
    const float* __restrict__ b_ih, const float* __restrict__ b_hh,
    float* __restrict__ deter_out) {
  int idx = blockIdx.x * 256 + threadIdx.x;  // 16*2048
  if (idx >= 16 * 2048) return;
  int h = idx & 2047;
  const float* g0 = ws + WS_GATE;
  float gr  = g0[idx] + g0[32768 + idx] + g0[65536 + idx] + g0[98304 + idx]
            + b_ih[h] + b_hh[h];
  float gz  = g0[131072 + idx] + g0[163840 + idx] + g0[196608 + idx] + g0[229376 + idx]
            + b_ih[2048 + h] + b_hh[2048 + h];
  float gin = g0[262144 + idx] + g0[294912 + idx] + b_ih[4096 + h];
  float ghn = g0[327680 + idx] + g0[360448 + idx] + b_hh[4096 + h];
  float r = 1.0f / (1.0f + expf(-gr));
  float z = 1.0f / (1.0f + expf(-gz));
  float n = tanhf(gin + r * ghn);
  deter_out[idx] = (1.0f - z) * n + z * prev_deter[idx];
}

// ---------------- hidden-layer partial GEMMs ----------------
// 640 waves: 0..255 prior (p=0,1), 256..639 posterior (p=0,1 deter; p=2 obs)
__global__ void __launch_bounds__(256) k_hidden_gemm(
    const float* __restrict__ deter, const float* __restrict__ observation,
    const float* __restrict__ pri_W1, const float* __restrict__ post_W1,
    float* __restrict__ ws) {
  const int gid  = blockIdx.x * 8 + (threadIdx.x >> 5);  // 0..639
  const int lane = threadIdx.x & 31;
  const int r16  = lane & 15;
  const int koff = (lane >> 4) * 2;

  const float* xp; const float* wp; float* dst; int hb;
  if (gid < 256) {
    int p = gid >> 7, tile = gid & 127;
    hb = tile * 16;
    xp = deter + r16 * 2048 + p * 1024 + koff;
    wp = pri_W1 + (size_t)(hb + r16) * 2048 + p * 1024 + koff;
    dst = ws + WS_HID + (size_t)p * 32768;
  } else {
    int j = gid - 256;
    int p = j >> 7, tile = j & 127;
    hb = tile * 16;
    if (p < 2) {
      xp = deter + r16 * 2048 + p * 1024 + koff;
      wp = post_W1 + (size_t)(hb + r16) * 3072 + p * 1024 + koff;
    } else {
      xp = observation + r16 * 1024 + koff;
      wp = post_W1 + (size_t)(hb + r16) * 3072 + 2048 + koff;
    }
    dst = ws + WS_HID + (size_t)(2 + p) * 32768;
  }

  v8f acc = {};
  acc = gemm_k<1024>(xp, wp, acc);
#pragma unroll
  for (int j = 0; j < 8; ++j) {
    int m = j + 8 * (lane >> 4);
    dst[m * 2048 + hb + r16] = acc[j];
  }
}

// ---------------- hidden activation: silu(sum + bias) ----------------
__global__ void __launch_bounds__(256) k_hidden_act(
    const float* __restrict__ ws_in,
    const float* __restrict__ pri_b1, const float* __restrict__ post_b1,
    float* __restrict__ h1_pri, float* __restrict__ h1_post) {
  int idx = blockIdx.x * 256 + threadIdx.x;  // 16*2048
  if (idx >= 16 * 2048) return;
  int h = idx & 2047;
  const float* hw = ws_in + WS_HID;
  float a = hw[idx] + hw[32768 + idx] + pri_b1[h];
  float b = hw[65536 + idx] + hw[98304 + idx] + hw[131072 + idx] + post_b1[h];
  h1_pri[idx]  = a / (1.0f + expf(-a));
  h1_post[idx] = b / (1.0f + expf(-b));
}

// ---------------- logit partial GEMMs ----------------
// 256 waves: half 0 prior, half 1 posterior; each half 2 K-parts x 64 tiles
__global__ void __launch_bounds__(256) k_logits_gemm(
    const float* __restrict__ h1_pri, const float* __restrict__ h1_post,
    const float* __restrict__ pri_W2, const float* __restrict__ post_W2,
    float* __restrict__ ws) {
  const int gid  = blockIdx.x * 8 + (threadIdx.x >> 5);  // 0..255
  const int lane = threadIdx.x & 31;
  const int r16  = lane & 15;
  const int koff = (lane >> 4) * 2;
  const int half = gid >> 7;
  const int j = gid & 127;
  const int p = j >> 6, tile = j & 63;
  const int sb = tile * 16;

  const float* h1 = half ? h1_post : h1_pri;
  const float* W  = half ? post_W2 : pri_W2;
  const float* xp = h1 + r16 * 2048 + p * 1024 + koff;
  const float* wp = W + (size_t)(sb + r16) * 2048 + p * 1024 + koff;

  v8f acc = {};
  acc = gemm_k<1024>(xp, wp, acc);

  float* dst = ws + WS_LOGIT + (size_t)half * 32768 + (size_t)p * 16384;
#pragma unroll
  for (int jj = 0; jj < 8; ++jj) {
    int m = jj + 8 * (lane >> 4);
    dst[m * 1024 + sb + r16] = acc[jj];
  }
}

// ---------------- logit finalize: sum partials + bias -> d_out ----------
__global__ void __launch_bounds__(256) k_logits_fin(
    const float* __restrict__ ws_in,
    const float* __restrict__ pri_b2, const float* __restrict__ post_b2,
    float* __restrict__ prior_out, float* __restrict__ post_out) {
  int idx = blockIdx.x * 256 + threadIdx.x;  // 2 * 16*1024
  if (idx >= 32768) return;
  const float* lw = ws_in + WS_LOGIT;
  if (idx < 16384) {
    prior_out[idx] = lw[idx] + lw[16384 + idx] + pri_b2[idx & 1023];
  } else {
    int i = idx - 16384;
    post_out[i] = lw[32768 + i] + lw[49152 + i] + post_b2[i & 1023];
  }
}

// ---------------- straight-through sample (stoch == onehot numerically) ---
static __device__ __forceinline__ unsigned mix32(unsigned x) {
  x ^= x >> 16; x *= 0x7feb352dU; x ^= x >> 15; x *= 0x846ca68bU; x ^= x >> 16;
  return x;
}
__global__ void __launch_bounds__(256) k_sample(
    const float* __restrict__ post_logits,  // [16,1024]
    float* __restrict__ stoch) {            // [16,1024]
  int t = blockIdx.x * 256 + threadIdx.x;   // one thread per (b,d) row
  if (t >= 512) return;
  int b = t >> 5, d = t & 31;
  const float* lg = post_logits + b * 1024 + d * 32;
  int best = 0; float bestv = -3.4e38f;
  for (int c = 0; c < 32; ++c) {
    unsigned u = mix32(mix32(42u ^ (unsigned)(t * 33 + c)) + 0x9e3779b9u);
    float uf = (float)((u >> 8) + 1u) * (1.0f / 16777217.0f);
    float g = -logf(-logf(uf));             // gumbel(0,1)
    float v = lg[c] + g;
    if (v > bestv) { bestv = v; best = c; }
  }
  float* out = stoch + b * 1024 + d * 32;
  for (int c = 0; c < 32; ++c) out[c] = (c == best) ? 1.0f : 0.0f;
}

extern "C" void kernel_launch(void* const* d_in, const int* in_sizes, int n_in,
                              void* d_out, int out_size, void* d_ws, size_t ws_size,
                              hipStream_t stream) {
  (void)in_sizes; (void)n_in; (void)out_size; (void)ws_size;
  const float* prev_stoch  = (const float*)d_in[0];
  const float* prev_deter  = (const float*)d_in[1];
  const float* action      = (const float*)d_in[2];
  const float* observation = (const float*)d_in[3];
  const float* W_ih   = (const float*)d_in[4];
  const float* W_hh   = (const float*)d_in[5];
  const float* b_ih   = (const float*)d_in[6];
  const float* b_hh   = (const float*)d_in[7];
  const float* pri_W1 = (const float*)d_in[8];
  const float* pri_b1 = (const float*)d_in[9];
  const float* pri_W2 = (const float*)d_in[10];
  const float* pri_b2 = (const float*)d_in[11];
  const float* post_W1 = (const float*)d_in[12];
  const float* post_b1 = (const float*)d_in[13];
  const float* post_W2 = (const float*)d_in[14];
  const float* post_b2 = (const float*)d_in[15];

  float* out = (float*)d_out;            // tuple flat: deter|stoch|prior|posterior
  float* deter_out = out;                // 16*2048
  float* stoch_out = out + 32768;        // 16*1024
  float* prior_out = out + 49152;        // 16*1024
  float* post_out  = out + 65536;        // 16*1024

  float* ws = (float*)d_ws;
  float* h1_pri  = ws + WS_H1PRI;
  float* h1_post = ws + WS_H1POST;

  k_gates_gemm<<<192, 256, 0, stream>>>(prev_stoch, prev_deter, action,
                                        W_ih, W_hh, ws);
  k_gru_combine<<<128, 256, 0, stream>>>(ws, prev_deter, b_ih, b_hh, deter_out);
  k_hidden_gemm<<<80, 256, 0, stream>>>(deter_out, observation,
                                        pri_W1, post_W1, ws);
  k_hidden_act<<<128, 256, 0, stream>>>(ws, pri_b1, post_b1, h1_pri, h1_post);
  k_logits_gemm<<<32, 256, 0, stream>>>(h1_pri, h1_post, pri_W2, post_W2, ws);
  k_logits_fin<<<128, 256, 0, stream>>>(ws, pri_b2, post_b2, prior_out, post_out);
  k_sample<<<2, 256, 0, stream>>>(post_out, stoch_out);
}